// R3NN_20117626814733
// MI455X (gfx1250) — compile-verified
//
#include <hip/hip_runtime.h>
#include <stdint.h>

// ---------------------------------------------------------------------------
// Reference is identity: out = tree  (8 x 4096 x 1024 fp32, 128 MiB).
// Pure streaming copy, 0 FLOPs -> HBM-bound (~11.5 us at 23.3 TB/s).
// Use the CDNA5 async global<->LDS engine (ASYNCcnt) with B128 transfers and
// double buffering so every wave overlaps one async load with one async store.
// ---------------------------------------------------------------------------

typedef __attribute__((ext_vector_type(4))) float fvec4;
typedef __attribute__((ext_vector_type(4))) int   ivec4;

#define GLOBAL_AS __attribute__((address_space(1)))
#define LDS_AS    __attribute__((address_space(3)))

#if defined(__HIP_DEVICE_COMPILE__) && defined(__gfx1250__) &&                 \
    __has_builtin(__builtin_amdgcn_global_load_async_to_lds_b128) &&           \
    __has_builtin(__builtin_amdgcn_global_store_async_from_lds_b128)
#define USE_ASYNC_LDS 1
#else
#define USE_ASYNC_LDS 0
#endif

__device__ __forceinline__ void wait_asynccnt0() {
#if defined(__HIP_DEVICE_COMPILE__) && defined(__gfx1250__)
#if __has_builtin(__builtin_amdgcn_s_wait_asynccnt)
  __builtin_amdgcn_s_wait_asynccnt(0);
#else
  asm volatile("s_wait_asynccnt 0" ::: "memory");
#endif
#endif
}

// One tile = 256 lanes x 16B = 4 KiB. Grid-stride over tiles.
__global__ void tree_identity_copy(const fvec4* __restrict__ src,
                                   fvec4* __restrict__ dst,
                                   int ntiles) {
#if USE_ASYNC_LDS
  __shared__ fvec4 buf[2][256];   // double buffer, 8 KiB of the 320 KiB WGP LDS
  const int tid = threadIdx.x;
  int t = (int)blockIdx.x;
  if (t >= ntiles) return;
  const int stride = (int)gridDim.x;

  LDS_AS ivec4* l0 = (LDS_AS ivec4*)&buf[0][tid];
  LDS_AS ivec4* l1 = (LDS_AS ivec4*)&buf[1][tid];

  // Prologue: async load tile t into buffer 0.
  __builtin_amdgcn_global_load_async_to_lds_b128(
      (GLOBAL_AS ivec4*)(src + (size_t)t * 256 + tid), l0,
      /*offset=*/0, /*cpol=*/0);

  int parity = 0;
  for (;;) {
    const int tn = t + stride;
    // Current load (and previous iteration's store) complete.
    wait_asynccnt0();
    if (tn < ntiles) {
      // Overlap: start next tile's load into the other buffer...
      __builtin_amdgcn_global_load_async_to_lds_b128(
          (GLOBAL_AS ivec4*)(src + (size_t)tn * 256 + tid),
          parity ? l0 : l1, 0, 0);
    }
    // ...while storing the current buffer out.
    __builtin_amdgcn_global_store_async_from_lds_b128(
        (GLOBAL_AS ivec4*)(dst + (size_t)t * 256 + tid),
        parity ? l1 : l0, 0, 0);
    if (tn >= ntiles) break;
    t = tn;
    parity ^= 1;
  }
  // Drain the final async store before the wave exits.
  wait_asynccnt0();
#else
  // Fallback: direct B128 copy with non-temporal hints (still bandwidth-optimal).
  const size_t n = (size_t)ntiles * 256;
  for (size_t i = (size_t)blockIdx.x * blockDim.x + threadIdx.x; i < n;
       i += (size_t)gridDim.x * blockDim.x) {
    __builtin_nontemporal_store(__builtin_nontemporal_load(&src[i]), &dst[i]);
  }
#endif
}

// Scalar tail for any remainder not covered by whole 4 KiB tiles
// (not hit for 2^25 elements, but keeps the launch fully general).
__global__ void tail_copy(const float* __restrict__ src, float* __restrict__ dst,
                          size_t n) {
  size_t i = (size_t)blockIdx.x * blockDim.x + threadIdx.x;
  if (i < n) dst[i] = src[i];
}

extern "C" void kernel_launch(void* const* d_in, const int* in_sizes, int n_in,
                              void* d_out, int out_size, void* d_ws, size_t ws_size,
                              hipStream_t stream) {
  (void)n_in; (void)d_ws; (void)ws_size;
  // Inputs in setup_inputs() order: 0=dsl(int), 1=embedded_samples(f32), 2=tree(f32).
  const float* tree = (const float*)d_in[2];
  float* out = (float*)d_out;

  const size_t n = (size_t)out_size;          // == in_sizes[2] == 33,554,432
  const size_t nvec = n / 4;                  // float4 count
  const int ntiles = (int)(nvec / 256);       // 4 KiB tiles

  if (ntiles > 0) {
    int grid = ntiles < 4096 ? ntiles : 4096; // ~8 tiles per block at full size
    tree_identity_copy<<<grid, 256, 0, stream>>>((const fvec4*)tree,
                                                 (fvec4*)out, ntiles);
  }
  const size_t done = (size_t)ntiles * 1024;  // floats already copied
  if (done < n) {
    const size_t rem = n - done;
    const int grid = (int)((rem + 255) / 256);
    tail_copy<<<grid, 256, 0, stream>>>(tree + done, out + done, rem);
  }
}